// CausalEmbeddingUpdater_51187420234436
// MI455X (gfx1250) — compile-verified
//
#include <hip/hip_runtime.h>
#include <hip/hip_bf16.h>
#include <stdint.h>
#include <stddef.h>

// Problem constants (from reference)
#define NN 100000
#define EE 800000
#define DHID 64
#define LLAYERS 2
#define SSAMP 4
#define NOISE_SCALE 0.1f
#define MIN_U 1e-6f

typedef float v2f __attribute__((ext_vector_type(2)));
typedef float v8f __attribute__((ext_vector_type(8)));

// ---------------------------------------------------------------------------
// helpers
// ---------------------------------------------------------------------------
__device__ __forceinline__ uint32_t hash_u32(uint32_t x) {
  x ^= x >> 17; x *= 0xed5ad4bbu;
  x ^= x >> 11; x *= 0xac4c1b51u;
  x ^= x >> 15; x *= 0x31848babu;
  x ^= x >> 14;
  return x;
}

// Guaranteed native non-returning fp32 atomic add at device scope
// (GLOBAL_ATOMIC_ADD_F32, tracked with STOREcnt; s_endpgm waits idle).
__device__ __forceinline__ void atomic_add_f32(float* p, float v) {
#if defined(__gfx1250__)
  asm volatile("global_atomic_add_f32 %0, %1, off scope:SCOPE_DEV"
               :: "v"((unsigned long long)(uintptr_t)p), "v"(v)
               : "memory");
#else
  __hip_atomic_fetch_add(p, v, __ATOMIC_RELAXED, __HIP_MEMORY_SCOPE_AGENT);
#endif
}

// ---------------------------------------------------------------------------
// Weight pre-swizzle: pack a 64x64 row-major W into per-fragment order so the
// GEMM's B-fragment load is a single aligned global_load_b64 per lane.
// Packed index p = ((kc*4 + t)*32 + lane)*2 + j  ->
//   row = kc*4 + ((lane>>4)<<1) + j , col = t*16 + (lane&15)
// ---------------------------------------------------------------------------
__global__ __launch_bounds__(256) void pack_w_kernel(
    const float* __restrict__ W, float* __restrict__ Wp) {
  const int p = blockIdx.x * blockDim.x + threadIdx.x;
  if (p >= DHID * DHID) return;
  const int j    = p & 1;
  const int lane = (p >> 1) & 31;
  const int t    = (p >> 6) & 3;
  const int kc   = p >> 8;
  const int row  = kc * 4 + ((lane >> 4) << 1) + j;
  const int col  = t * 16 + (lane & 15);
  Wp[p] = W[row * DHID + col];
}

// ---------------------------------------------------------------------------
// WMMA GEMM: out[rows,64] = act( X1[rows,64] @ W1 (+ X2 @ W2) + bias )
// W1/W2 are pre-packed (fragment order). One wave = 16 rows x 64 cols.
// MODE 0: store result.  MODE 1: sumz += result; ss[row] += sum_col(result^2)
// ---------------------------------------------------------------------------
template <int NACC, bool RELU, int MODE>
__global__ __launch_bounds__(256) void gemm64_kernel(
    const float* __restrict__ X1, const float* __restrict__ W1p,
    const float* __restrict__ X2, const float* __restrict__ W2p,
    const float* __restrict__ bias,
    float* __restrict__ out, float* __restrict__ ss, int nrows) {
  const int lane = threadIdx.x & 31;
  const int wave = blockIdx.x * (blockDim.x >> 5) + (threadIdx.x >> 5);
  const int row0 = wave * 16;
  if (row0 >= nrows) return;  // wave-uniform: EXEC stays all-ones for WMMA

  const int m    = lane & 15;          // M index (A) / N index (B,C)
  const int koff = (lane >> 4) << 1;   // 0 for lanes 0-15, 2 for lanes 16-31
  const int rhi  = (lane >> 4) << 3;   // row offset for C (0 or 8)

  v8f c0 = {}, c1 = {}, c2 = {}, c3 = {};

  // ---- accumulate X1 @ W1 ----
  {
    const float* Xrow = X1 + (size_t)(row0 + m) * DHID;
    const v2f* Wv = (const v2f*)W1p;
#pragma unroll
    for (int kc = 0; kc < 16; ++kc) {
      const int kb = kc * 4 + koff;
      v2f a;
      a.x = Xrow[kb];
      a.y = Xrow[kb + 1];
      const int bidx = kc * 128 + lane;
      v2f b0 = Wv[bidx];
      v2f b1 = Wv[bidx + 32];
      v2f b2 = Wv[bidx + 64];
      v2f b3 = Wv[bidx + 96];
      c0 = __builtin_amdgcn_wmma_f32_16x16x4_f32(false, a, false, b0, (short)0, c0, false, false);
      c1 = __builtin_amdgcn_wmma_f32_16x16x4_f32(false, a, false, b1, (short)0, c1, false, false);
      c2 = __builtin_amdgcn_wmma_f32_16x16x4_f32(false, a, false, b2, (short)0, c2, false, false);
      c3 = __builtin_amdgcn_wmma_f32_16x16x4_f32(false, a, false, b3, (short)0, c3, false, false);
    }
  }
  // ---- optional second accumulation X2 @ W2 ----
  if (NACC == 2) {
    const float* Xrow = X2 + (size_t)(row0 + m) * DHID;
    const v2f* Wv = (const v2f*)W2p;
#pragma unroll
    for (int kc = 0; kc < 16; ++kc) {
      const int kb = kc * 4 + koff;
      v2f a;
      a.x = Xrow[kb];
      a.y = Xrow[kb + 1];
      const int bidx = kc * 128 + lane;
      v2f b0 = Wv[bidx];
      v2f b1 = Wv[bidx + 32];
      v2f b2 = Wv[bidx + 64];
      v2f b3 = Wv[bidx + 96];
      c0 = __builtin_amdgcn_wmma_f32_16x16x4_f32(false, a, false, b0, (short)0, c0, false, false);
      c1 = __builtin_amdgcn_wmma_f32_16x16x4_f32(false, a, false, b1, (short)0, c1, false, false);
      c2 = __builtin_amdgcn_wmma_f32_16x16x4_f32(false, a, false, b2, (short)0, c2, false, false);
      c3 = __builtin_amdgcn_wmma_f32_16x16x4_f32(false, a, false, b3, (short)0, c3, false, false);
    }
  }

  // ---- epilogue ----
  float sq[8] = {0.f, 0.f, 0.f, 0.f, 0.f, 0.f, 0.f, 0.f};
#pragma unroll
  for (int t = 0; t < 4; ++t) {
    v8f c = (t == 0) ? c0 : (t == 1) ? c1 : (t == 2) ? c2 : c3;
    const int col = t * 16 + m;
    const float bv = bias ? bias[col] : 0.0f;
#pragma unroll
    for (int v = 0; v < 8; ++v) {
      const int row = row0 + v + rhi;
      float val = c[v] + bv;
      if (RELU) val = fmaxf(val, 0.0f);
      const size_t o = (size_t)row * DHID + col;
      if (MODE == 0) {
        out[o] = val;
      } else {
        out[o] += val;          // sumz accumulation (one thread per element)
        sq[v] += val * val;
      }
    }
  }
  if (MODE == 1) {
#pragma unroll
    for (int v = 0; v < 8; ++v) {
      float s = sq[v];
      s += __shfl_xor(s, 1, 16);
      s += __shfl_xor(s, 2, 16);
      s += __shfl_xor(s, 4, 16);
      s += __shfl_xor(s, 8, 16);
      if (m == 0) ss[row0 + v + rhi] += s;  // row owned by exactly one wave
    }
  }
}

// ---------------------------------------------------------------------------
// edge-degree count:  cnt[dst[e]] += mask[e]
// ---------------------------------------------------------------------------
__global__ __launch_bounds__(256) void edge_count_kernel(
    const int* __restrict__ dst, const unsigned char* __restrict__ mask,
    float* __restrict__ cnt, int E) {
  int e = blockIdx.x * blockDim.x + threadIdx.x;
  if (e >= E) return;
  if (mask[e]) atomic_add_f32(&cnt[dst[e]], 1.0f);
}

// cnt_inv = 1/max(cnt,1) ; ind = (cnt>0)
__global__ __launch_bounds__(256) void cnt_finalize_kernel(
    const float* __restrict__ cnt, float* __restrict__ cnt_inv,
    float* __restrict__ ind, int n) {
  int i = blockIdx.x * blockDim.x + threadIdx.x;
  if (i >= n) return;
  const float c = cnt[i];
  cnt_inv[i] = 1.0f / fmaxf(c, 1.0f);
  ind[i] = (c > 0.0f) ? 1.0f : 0.0f;
}

// ---------------------------------------------------------------------------
// masked scatter-add of B[src] into agg[dst]; ONE WAVE PER EDGE:
// lane covers 2 features (v2f gather, 256B contiguous per edge; B and agg are
// L2-resident at 25.6MB). mask/src/dst are wave-uniform scalar loads and the
// masked-out 30% of edges exit wave-uniformly.
// ---------------------------------------------------------------------------
__global__ __launch_bounds__(256) void edge_scatter_kernel(
    const int* __restrict__ src, const int* __restrict__ dst,
    const unsigned char* __restrict__ mask, const float* __restrict__ B,
    float* __restrict__ agg, int E) {
  const int lane = threadIdx.x & 31;
  int e = blockIdx.x * (blockDim.x >> 5) + (threadIdx.x >> 5);
  if (e >= E) return;                      // wave-uniform
  e = __builtin_amdgcn_readfirstlane(e);   // force scalar addressing
  if (!mask[e]) return;                    // wave-uniform exit
  const int s = __builtin_amdgcn_readfirstlane(src[e]);
  const int t = __builtin_amdgcn_readfirstlane(dst[e]);
  const v2f v = *(const v2f*)(B + (size_t)s * DHID + lane * 2);
  float* ap = agg + (size_t)t * DHID + lane * 2;
  atomic_add_f32(ap, v.x);
  atomic_add_f32(ap + 1, v.y);
}

// agg = ind[n]*A + cnt_inv[n]*agg   (in place on the raw scatter buffer)
__global__ __launch_bounds__(256) void agg_finalize_kernel(
    const float* __restrict__ A, float* __restrict__ agg,
    const float* __restrict__ cnt_inv, const float* __restrict__ ind,
    int total) {
  int i = blockIdx.x * blockDim.x + threadIdx.x;
  if (i >= total) return;
  const int n = i >> 6;
  agg[i] = ind[n] * A[i] + cnt_inv[n] * agg[i];
}

// xin = x + NOISE_SCALE * N(0,1)   (deterministic hash-based Box-Muller)
__global__ __launch_bounds__(256) void add_noise_kernel(
    const float* __restrict__ x, float* __restrict__ xin, int total,
    uint32_t seed) {
  int i = blockIdx.x * blockDim.x + threadIdx.x;
  if (i >= total) return;
  const uint32_t h1 = hash_u32((uint32_t)i * 2u + seed);
  const uint32_t h2 = hash_u32((uint32_t)i * 2u + 1u + seed);
  const float u1 = ((float)h1 + 1.0f) * 2.3283064e-10f;  // (0,1]
  const float u2 = (float)h2 * 2.3283064e-10f;           // [0,1)
  const float g = sqrtf(-2.0f * __logf(u1)) * __cosf(6.28318530718f * u2);
  xin[i] = x[i] + NOISE_SCALE * g;
}

// u[n] = max( (ss[n] - ||sumz[n]||^2/S) / (S-1), MIN_U )   one wave per node
__global__ __launch_bounds__(256) void uncertainty_kernel(
    const float* __restrict__ sumz, const float* __restrict__ ss,
    float* __restrict__ u, int n) {
  const int lane = threadIdx.x & 31;
  const int node = blockIdx.x * (blockDim.x >> 5) + (threadIdx.x >> 5);
  if (node >= n) return;
  float acc = 0.0f;
  for (int i = lane; i < DHID; i += 32) {
    const float v = sumz[(size_t)node * DHID + i];
    acc += v * v;
  }
  acc += __shfl_xor(acc, 1);
  acc += __shfl_xor(acc, 2);
  acc += __shfl_xor(acc, 4);
  acc += __shfl_xor(acc, 8);
  acc += __shfl_xor(acc, 16);
  if (lane == 0) {
    float val = (ss[node] - acc / (float)SSAMP) / (float)(SSAMP - 1);
    u[node] = fmaxf(val, MIN_U);
  }
}

// ---------------------------------------------------------------------------
// one full GNN evaluation (weights pre-packed; wp = packed base)
// matrix slots: 0=Wi, 1+2l=msgTop[l], 2+2l=msgBot[l], 5+2l=updTop[l],
//               6+2l=updBot[l], 9=Wf
// ---------------------------------------------------------------------------
static void run_gnn(const float* X, const float* wp,
                    const float* bi, const float* msgb, const float* updb,
                    const float* bf,
                    const int* src, const int* dst, const unsigned char* mask,
                    float* h, float* A, float* B, float* agg,
                    const float* cnt_inv, const float* ind,
                    float* zout, float* sumz, float* ssbuf, bool causal,
                    hipStream_t stream) {
  const int ggrid = (NN / 16 + 7) / 8;   // 16-row waves, 8 waves / block
  const int egrid = (EE + 7) / 8;        // 1 wave per edge, 8 waves / block
  const int ngrid = (NN * DHID + 255) / 256;
  const int WSZ = DHID * DHID;

  // h = relu(x @ Wi + bi)
  gemm64_kernel<1, true, 0><<<ggrid, 256, 0, stream>>>(
      X, wp + 0 * WSZ, nullptr, nullptr, bi, h, nullptr, NN);

  for (int l = 0; l < LLAYERS; ++l) {
    // A = h @ Wm_top + msg_b ; B = h @ Wm_bot
    gemm64_kernel<1, false, 0><<<ggrid, 256, 0, stream>>>(
        h, wp + (1 + 2 * l) * WSZ, nullptr, nullptr,
        msgb + (size_t)l * DHID, A, nullptr, NN);
    gemm64_kernel<1, false, 0><<<ggrid, 256, 0, stream>>>(
        h, wp + (2 + 2 * l) * WSZ, nullptr, nullptr, nullptr, B, nullptr, NN);
    // agg = ind*A + cnt_inv * scatter_sum(mask * B[src], dst)
    hipMemsetAsync(agg, 0, (size_t)NN * DHID * sizeof(float), stream);
    edge_scatter_kernel<<<egrid, 256, 0, stream>>>(src, dst, mask, B, agg, EE);
    agg_finalize_kernel<<<ngrid, 256, 0, stream>>>(A, agg, cnt_inv, ind,
                                                   NN * DHID);
    // h = relu(h @ Wu_top + agg @ Wu_bot + upd_b)  (in place: rows wave-owned)
    gemm64_kernel<2, true, 0><<<ggrid, 256, 0, stream>>>(
        h, wp + (5 + 2 * l) * WSZ, agg, wp + (6 + 2 * l) * WSZ,
        updb + (size_t)l * DHID, h, nullptr, NN);
  }

  if (causal) {
    gemm64_kernel<1, false, 0><<<ggrid, 256, 0, stream>>>(
        h, wp + 9 * WSZ, nullptr, nullptr, bf, zout, nullptr, NN);
  } else {
    gemm64_kernel<1, false, 1><<<ggrid, 256, 0, stream>>>(
        h, wp + 9 * WSZ, nullptr, nullptr, bf, sumz, ssbuf, NN);
  }
}

// ---------------------------------------------------------------------------
// entry point
// ---------------------------------------------------------------------------
extern "C" void kernel_launch(void* const* d_in, const int* in_sizes, int n_in,
                              void* d_out, int out_size, void* d_ws,
                              size_t ws_size, hipStream_t stream) {
  (void)in_sizes; (void)n_in; (void)out_size; (void)ws_size;
  const float* x    = (const float*)d_in[0];
  const int* eidx   = (const int*)d_in[1];
  const unsigned char* mask = (const unsigned char*)d_in[2];
  const float* Wi   = (const float*)d_in[3];
  const float* bi   = (const float*)d_in[4];
  const float* msgW = (const float*)d_in[5];
  const float* msgb = (const float*)d_in[6];
  const float* updW = (const float*)d_in[7];
  const float* updb = (const float*)d_in[8];
  const float* Wf   = (const float*)d_in[9];
  const float* bf   = (const float*)d_in[10];

  const int* src = eidx;
  const int* dst = eidx + EE;

  // workspace carve-out
  const size_t nd = (size_t)NN * DHID;
  const int WSZ = DHID * DHID;
  float* h      = (float*)d_ws;
  float* A      = h + nd;
  float* B      = A + nd;
  float* agg    = B + nd;
  float* xin    = agg + nd;
  float* sumz   = xin + nd;
  float* cnt    = sumz + nd;
  float* cntinv = cnt + NN;
  float* ind    = cntinv + NN;
  float* ssbuf  = ind + NN;
  float* wp     = ssbuf + NN;     // 10 packed 64x64 weight matrices (160 KB)

  float* out_z = (float*)d_out;   // [N,64]
  float* out_u = out_z + nd;      // [N]

  // --- pack all weight matrices into fragment order (once per launch) ---
  const float* wsrc[10] = {Wi,
                           msgW + 0 * WSZ, msgW + 1 * WSZ,
                           msgW + 2 * WSZ, msgW + 3 * WSZ,
                           updW + 0 * WSZ, updW + 1 * WSZ,
                           updW + 2 * WSZ, updW + 3 * WSZ,
                           Wf};
  // reorder msg/upd: slot layout is 0=Wi, 1..4 = msgTop0,msgBot0,msgTop1,msgBot1,
  // 5..8 = updTop0,updBot0,updTop1,updBot1, 9=Wf. msg_W[l] is (128,64): top
  // half = first 64 rows, bottom = next 64 rows, so l-major order already
  // matches (l*2 + half).
  for (int i = 0; i < 10; ++i)
    pack_w_kernel<<<(WSZ + 255) / 256, 256, 0, stream>>>(wsrc[i],
                                                         wp + (size_t)i * WSZ);

  // --- degree counts (shared across all runs) + stat accumulators ---
  hipMemsetAsync(cnt, 0, (size_t)NN * sizeof(float), stream);
  hipMemsetAsync(sumz, 0, nd * sizeof(float), stream);
  hipMemsetAsync(ssbuf, 0, (size_t)NN * sizeof(float), stream);
  edge_count_kernel<<<(EE + 255) / 256, 256, 0, stream>>>(dst, mask, cnt, EE);
  cnt_finalize_kernel<<<(NN + 255) / 256, 256, 0, stream>>>(cnt, cntinv, ind,
                                                            NN);

  // --- causal run ---
  run_gnn(x, wp, bi, msgb, updb, bf, src, dst, mask,
          h, A, B, agg, cntinv, ind, out_z, sumz, ssbuf, true, stream);

  // --- Monte Carlo samples ---
  for (int s = 0; s < SSAMP; ++s) {
    add_noise_kernel<<<(int)((nd + 255) / 256), 256, 0, stream>>>(
        x, xin, (int)nd, 0x9E3779B9u * (uint32_t)(s + 1));
    run_gnn(xin, wp, bi, msgb, updb, bf, src, dst, mask,
            h, A, B, agg, cntinv, ind, nullptr, sumz, ssbuf, false, stream);
  }

  // --- uncertainty ---
  uncertainty_kernel<<<(NN * 32 + 255) / 256, 256, 0, stream>>>(sumz, ssbuf,
                                                                out_u, NN);
}